// NUPointNetEncoder_15556371546707
// MI455X (gfx1250) — compile-verified
//
#include <hip/hip_runtime.h>
#include <hip/hip_bf16.h>
#include <math.h>

typedef float v2f __attribute__((ext_vector_type(2)));
typedef float v4f __attribute__((ext_vector_type(4)));
typedef float v8f __attribute__((ext_vector_type(8)));

#define BATCH 2
#define NPTS  8192
#define KNN   16
#define CCH   64
#define PEDIM 32
#define HIDD  16

// ---------------------------------------------------------------------------
// Kernel 1: h[b,o,n] = relu( sum_c W1[o,c] * f[b,c,n] + b1[o] )
// One wave per 16x16 output tile, K=64 via 16 x V_WMMA_F32_16X16X4_F32 (exact f32).
// Tiles: B(2) * Mtiles(4) * Ntiles(512) = 4096 waves; 4 waves / 128-thread block.
// ---------------------------------------------------------------------------
__global__ __launch_bounds__(128) void conv1x1_wmma(
    const float* __restrict__ f, const float* __restrict__ W1,
    const float* __restrict__ b1, float* __restrict__ h)
{
    const int wave = threadIdx.x >> 5;
    const int lane = threadIdx.x & 31;
    const int tile = blockIdx.x * 4 + wave;      // 0..4095
    const int ntile = tile & 511;
    const int mtile = (tile >> 9) & 3;
    const int b     = tile >> 11;
    const int half  = lane >> 4;                 // 0: lanes 0-15, 1: lanes 16-31
    const int l     = lane & 15;

    const float* Wrow  = W1 + (size_t)(mtile * 16 + l) * CCH;          // A row M=l
    const float* fbase = f + (size_t)b * CCH * NPTS + ntile * 16 + l;  // B col N=l

    v8f acc = {};
#pragma unroll
    for (int kk = 0; kk < 16; ++kk) {
        const int k0 = kk * 4 + 2 * half;        // lanes 0-15: K={0,1}; 16-31: K={2,3}
        v2f a, bm;
        a.x  = Wrow[k0];
        a.y  = Wrow[k0 + 1];
        bm.x = fbase[(size_t)k0 * NPTS];
        bm.y = fbase[(size_t)(k0 + 1) * NPTS];
        acc = __builtin_amdgcn_wmma_f32_16x16x4_f32(
            /*neg_a=*/false, a, /*neg_b=*/false, bm,
            /*c_mod=*/(short)0, acc, /*reuse_a=*/false, /*reuse_b=*/false);
    }

    // C/D layout: VGPR v -> row M = v + 8*half, col N = lane&15
    float* hb = h + (size_t)b * CCH * NPTS + ntile * 16 + l;
#pragma unroll
    for (int v = 0; v < 8; ++v) {
        const int m = mtile * 16 + v + 8 * half;
        const float val = acc[v] + b1[m];
        hb[(size_t)m * NPTS] = fmaxf(val, 0.0f);
    }
}

// ---------------------------------------------------------------------------
// Kernel 2: fout[b,c,n] = max_k( pe[b,c,n,k] + h[b,c,knn[b,n,k]] )
// Block = 256 threads = 64 channels x 4 points; knn staged once in LDS.
// pe streamed with nontemporal b128 loads (67 MB, the dominant traffic).
// ---------------------------------------------------------------------------
__global__ __launch_bounds__(256) void fout_kernel(
    const float* __restrict__ pe, const float* __restrict__ h,
    const int* __restrict__ knn, float* __restrict__ fout)
{
    const int blk = blockIdx.x;                  // B * N/4 = 4096 blocks
    const int b   = blk >> 11;
    const int n0  = (blk & 2047) * 4;

    __shared__ int sidx[4 * KNN];
    if (threadIdx.x < 64)
        sidx[threadIdx.x] = knn[((size_t)b * NPTS + n0) * KNN + threadIdx.x];
    __syncthreads();

    const int dn = threadIdx.x & 3;
    const int c  = threadIdx.x >> 2;
    const int n  = n0 + dn;

    const v4f*   pe4  = (const v4f*)(pe + (((size_t)b * CCH + c) * NPTS + n) * KNN);
    const float* hrow = h + ((size_t)b * CCH + c) * NPTS;
    const int*   idx  = &sidx[dn * KNN];

    float m = -INFINITY;
#pragma unroll
    for (int q = 0; q < 4; ++q) {
        const v4f pv = __builtin_nontemporal_load(&pe4[q]);
        m = fmaxf(m, pv.x + hrow[idx[q * 4 + 0]]);
        m = fmaxf(m, pv.y + hrow[idx[q * 4 + 1]]);
        m = fmaxf(m, pv.z + hrow[idx[q * 4 + 2]]);
        m = fmaxf(m, pv.w + hrow[idx[q * 4 + 3]]);
    }
    fout[((size_t)b * CCH + c) * NPTS + n] = m;
}

// ---------------------------------------------------------------------------
// Kernel 3: full KDE offset head. One wave per point (lane == PE dim).
// 8 points per 256-thread block; grid = 16384/8 = 2048.
// ---------------------------------------------------------------------------
__global__ __launch_bounds__(256) void kde_kernel(
    const float* __restrict__ p, const int* __restrict__ knn,
    const float* __restrict__ Wpe, const float* __restrict__ bpe,
    const float* __restrict__ Wh1, const float* __restrict__ bh1,
    const float* __restrict__ Wh2, const float* __restrict__ bh2,
    float* __restrict__ delta)
{
    __shared__ float sWpe[PEDIM * 3], sbpe[PEDIM];
    __shared__ float sWh1[HIDD * PEDIM], sbh1[HIDD];
    __shared__ float sWh2[3 * HIDD], sbh2[3];
    __shared__ float scx[8][KNN], scy[8][KNN], scz[8][KNN];
    __shared__ float skde[8][KNN];
    __shared__ float sfused[8][KNN][PEDIM];
    __shared__ float st[8][KNN][HIDD];

    const int tid = threadIdx.x;
    for (int i = tid; i < PEDIM * 3;    i += 256) sWpe[i] = Wpe[i];
    for (int i = tid; i < PEDIM;        i += 256) sbpe[i] = bpe[i];
    for (int i = tid; i < HIDD * PEDIM; i += 256) sWh1[i] = Wh1[i];
    for (int i = tid; i < HIDD;         i += 256) sbh1[i] = bh1[i];
    for (int i = tid; i < 3 * HIDD;     i += 256) sWh2[i] = Wh2[i];
    for (int i = tid; i < 3;            i += 256) sbh2[i] = bh2[i];

    const int wave  = tid >> 5;
    const int lane  = tid & 31;
    const int point = blockIdx.x * 8 + wave;     // 0..16383 == b*N + n
    const int b     = point >> 13;
    const int* idx  = knn + (size_t)point * KNN;

    // stage neighbor coordinates (lanes 0..15 -> neighbor k)
    if (lane < KNN) {
        const int j = idx[lane];
        const float* pp = p + ((size_t)b * NPTS + j) * 3;
        scx[wave][lane] = pp[0];
        scy[wave][lane] = pp[1];
        scz[wave][lane] = pp[2];
    }
    __syncthreads();

    // kde[k] = coef * sum_j exp(-d2(k,j)/(2s^2+eps)); normalize by max_k
    const float coef   = rsqrtf(2.0f * (float)M_PI * 1.0f + 1e-6f);
    const float invden = 1.0f / (2.0f * 1.0f + 1e-6f);
    if (lane < KNN) {
        const float xi = scx[wave][lane], yi = scy[wave][lane], zi = scz[wave][lane];
        float s = 0.0f;
#pragma unroll
        for (int j = 0; j < KNN; ++j) {
            const float dx = xi - scx[wave][j];
            const float dy = yi - scy[wave][j];
            const float dz = zi - scz[wave][j];
            const float d2 = dx * dx + dy * dy + dz * dz;
            s += expf(-d2 * invden);
        }
        s *= coef;
        float mx = s;
#pragma unroll
        for (int off = 8; off >= 1; off >>= 1)
            mx = fmaxf(mx, __shfl_xor(mx, off, 32));
        skde[wave][lane] = s / (mx + 1e-6f);
    }
    __syncthreads();

    // pek[k][d] = mean_j relu(Wpe[d]·rel[k,j] + bpe[d]);  fused = pek * kde[k]
    {
        const int d = lane;                      // PE_DIM == warp width
        const float w0 = sWpe[d * 3 + 0], w1 = sWpe[d * 3 + 1], w2 = sWpe[d * 3 + 2];
        const float bb = sbpe[d];
#pragma unroll
        for (int k = 0; k < KNN; ++k) {
            const float xk = scx[wave][k], yk = scy[wave][k], zk = scz[wave][k];
            float acc = 0.0f;
#pragma unroll
            for (int j = 0; j < KNN; ++j) {
                const float rx = xk - scx[wave][j];
                const float ry = yk - scy[wave][j];
                const float rz = zk - scz[wave][j];
                acc += fmaxf(fmaf(w0, rx, fmaf(w1, ry, fmaf(w2, rz, bb))), 0.0f);
            }
            sfused[wave][k][d] = (acc * (1.0f / 16.0f)) * skde[wave][k];
        }
    }
    __syncthreads();

    // t[k][hh] = relu(fused[k]·Wh1[hh] + bh1[hh]); lanes 0-15: hh 0..7, 16-31: hh 8..15
    {
        const int k     = lane & 15;
        const int hbase = (lane >> 4) * 8;
#pragma unroll
        for (int hh = hbase; hh < hbase + 8; ++hh) {
            float acc = sbh1[hh];
#pragma unroll
            for (int d = 0; d < PEDIM; ++d)
                acc = fmaf(sfused[wave][k][d], sWh1[hh * PEDIM + d], acc);
            st[wave][k][hh] = fmaxf(acc, 0.0f);
        }
    }
    __syncthreads();

    // delta[k] = t[k] @ Wh2^T + bh2 ; lanes 0-15 -> k
    if (lane < KNN) {
        float o0 = sbh2[0], o1 = sbh2[1], o2 = sbh2[2];
#pragma unroll
        for (int hh = 0; hh < HIDD; ++hh) {
            const float t = st[wave][lane][hh];
            o0 = fmaf(t, sWh2[0 * HIDD + hh], o0);
            o1 = fmaf(t, sWh2[1 * HIDD + hh], o1);
            o2 = fmaf(t, sWh2[2 * HIDD + hh], o2);
        }
        float* out = delta + ((size_t)point * KNN + lane) * 3;
        out[0] = o0; out[1] = o1; out[2] = o2;
    }
}

// ---------------------------------------------------------------------------
extern "C" void kernel_launch(void* const* d_in, const int* in_sizes, int n_in,
                              void* d_out, int out_size, void* d_ws, size_t ws_size,
                              hipStream_t stream) {
    const float* p   = (const float*)d_in[0];
    const float* f   = (const float*)d_in[1];
    const float* pe  = (const float*)d_in[2];
    const float* W1  = (const float*)d_in[3];
    const float* b1  = (const float*)d_in[4];
    const float* Wpe = (const float*)d_in[5];
    const float* bpe = (const float*)d_in[6];
    const float* Wh1 = (const float*)d_in[7];
    const float* bh1 = (const float*)d_in[8];
    const float* Wh2 = (const float*)d_in[9];
    const float* bh2 = (const float*)d_in[10];
    const int*   knn = (const int*)d_in[11];

    float* fout  = (float*)d_out;                       // [B,C,N] = 1,048,576 f32
    float* delta = fout + (size_t)BATCH * CCH * NPTS;   // [B*N,K,3] = 786,432 f32
    float* h     = (float*)d_ws;                        // [B,C,N] scratch, 4 MB

    conv1x1_wmma<<<1024, 128, 0, stream>>>(f, W1, b1, h);
    fout_kernel<<<BATCH * NPTS / 4, 256, 0, stream>>>(pe, h, knn, fout);
    kde_kernel<<<BATCH * NPTS / 8, 256, 0, stream>>>(p, knn, Wpe, bpe, Wh1, bh1,
                                                     Wh2, bh2, delta);
}